// ConvKernel_86870008529516
// MI455X (gfx1250) — compile-verified
//
#include <hip/hip_runtime.h>
#include <stdint.h>

typedef float  v2f  __attribute__((ext_vector_type(2)));
typedef float  v8f  __attribute__((ext_vector_type(8)));
typedef unsigned int u32x4 __attribute__((ext_vector_type(4)));
typedef int    i32x4 __attribute__((ext_vector_type(4)));
typedef int    i32x8 __attribute__((ext_vector_type(8)));

// ---- problem constants (from the reference) ----
#define Bn   2
#define Cn   16
#define D1n  8
#define D2n  8
#define D3n  64
#define D4n  64
#define On   32
#define KTOT 1296          // Cn * 3^4 reduction depth
#define KC   108           // K-chunk staged in LDS (1296 = 12*108)
#define NCHUNK 12
#define STEPS 27           // 108 / 4 k-values per f32 WMMA

// input strides (in floats): x[B][C][D1][D2][D3][D4]
#define XSC  (D1n*D2n*D3n*D4n)   // 262144
#define XS1  (D2n*D3n*D4n)       // 32768
#define XS2  (D3n*D4n)           // 4096
#define XS3  (D4n)               // 64
#define XSB  (Cn*XSC)            // 4194304

// output: out[B][O][6][6][62][62]
#define D1o 6
#define D2o 6
#define D3o 62
#define D4o 62
#define OS3 (D4o)            // 62
#define OS2 (D3o*D4o)        // 3844
#define OS1 (D2o*OS2)        // 23064
#define OSO (D1o*OS1)        // 138384
#define OSB (On*OSO)         // 4428288

#define ROWS (Bn*D1o*D2o*D3o)  // 4464 GEMM "rows" of 62 d4-outputs each

// ---- TDM: DMA one 108(k) x 32(o) weight tile (row stride 1296) into LDS ----
__device__ __forceinline__ void tdm_load_w_chunk(const float* w, int chunk,
                                                 unsigned lds_byte_off) {
  unsigned long long ga = (unsigned long long)(uintptr_t)w
                        + (unsigned long long)chunk * (KC * 4u);
  u32x4 g0;
  g0[0] = 1u;                                   // count=1 (valid user D#)
  g0[1] = lds_byte_off;                         // lds_addr (bytes)
  g0[2] = (unsigned)ga;                         // global_addr[31:0]
  g0[3] = (unsigned)((ga >> 32) & 0x01FFFFFFull) | (2u << 30); // [56:32] | type=2
  i32x8 g1;
  g1[0] = (int)(2u << 16);                      // data_size=4B; no multicast/pad
  g1[1] = (int)((unsigned)KTOT << 16);          // tensor_dim0[15:0]<<16
  g1[2] = (int)((unsigned)On << 16);            // tensor_dim1[15:0]<<16 | td0[31:16]=0
  g1[3] = (int)((unsigned)KC << 16);            // tile_dim0<<16 | td1[31:16]=0
  g1[4] = On;                                   // tile_dim2=0 <<16 | tile_dim1=32
  g1[5] = KTOT;                                 // tensor_dim0_stride[31:0] = 1296
  g1[6] = 0;
  g1[7] = 0;
  i32x4 z4 = {0,0,0,0};
#if __clang_major__ >= 23
  i32x8 z8 = {0,0,0,0,0,0,0,0};
  __builtin_amdgcn_tensor_load_to_lds(g0, g1, z4, z4, z8, 0);
#else
  __builtin_amdgcn_tensor_load_to_lds(g0, g1, z4, z4, 0);
#endif
}

__launch_bounds__(256)
__global__ void conv4d_relu_wmma(const float* __restrict__ x,
                                 const float* __restrict__ w,
                                 float* __restrict__ out) {
  __shared__ __align__(16) unsigned offs[KTOT];     // k -> input BYTE offset table
  __shared__ __align__(16) float bsh[2][On * KC];   // double-buffered weights [o][k]

  const int tid   = threadIdx.x;
  const int lane  = tid & 31;        // wave32
  const int wave  = tid >> 5;        // 8 waves per workgroup
  const int laneM = lane & 15;
  const int laneH = lane >> 4;       // 0: K0/K1 half, 1: K2/K3 half

  // One-time: byte offset of reduction term k inside the input tensor.
  // Whole input is 33.5 MB, so every address fits an unsigned 32-bit byte
  // offset -> enables saddr(x) + zext(vgpr32) global-load addressing.
  for (int i = tid; i < KTOT; i += 256) {
    int c  = i / 81;  int r = i - c * 81;
    int k1 = r / 27;  r -= k1 * 27;
    int k2 = r / 9;   r -= k2 * 9;
    int k3 = r / 3;   int k4 = r - k3 * 3;
    offs[i] = (unsigned)(c * XSC + k1 * XS1 + k2 * XS2 + k3 * XS3 + k4) * 4u;
  }

  if (tid == 0) tdm_load_w_chunk(w, 0, (unsigned)(uintptr_t)&bsh[0][0]);
  __syncthreads();   // offs table ready for everyone

  // Per-wave output tile: row = (b,d1,d2,d3), 16 consecutive d4 outputs.
  const int row = blockIdx.x * 2 + (wave >> 2);   // 2 rows per workgroup
  const int mt  = wave & 3;                       // 4 d4-tiles cover 62 cols
  int r = row;
  const int b  = r / (D1o * D2o * D3o); r -= b * (D1o * D2o * D3o);
  const int d1 = r / (D2o * D3o);       r -= d1 * (D2o * D3o);
  const int d2 = r / D3o;
  const int d3 = r - d2 * D3o;
  const int d4base = mt * 16;
  int d4 = d4base + laneM;                 // clamp tail lanes: keeps every
  if (d4 > D4o - 1) d4 = D4o - 1;          // gather in-bounds, EXEC stays all-1

  // Per-lane 32-bit byte offset of A row base (b, d1, d2, d3, d4).
  const unsigned abyte =
      ((unsigned)b * XSB + (unsigned)(d1 * XS1 + d2 * XS2 + d3 * XS3 + d4)) * 4u;
  const char* xb = (const char*)x;

  v8f acc0 = {0,0,0,0,0,0,0,0};   // N = 0..15
  v8f acc1 = {0,0,0,0,0,0,0,0};   // N = 16..31

  for (int ch = 0; ch < NCHUNK; ++ch) {
    if (tid == 0) __builtin_amdgcn_s_wait_tensorcnt(0);   // chunk ch landed
    __syncthreads();
    if (tid == 0 && ch + 1 < NCHUNK)                      // prefetch next chunk
      tdm_load_w_chunk(w, ch + 1, (unsigned)(uintptr_t)&bsh[(ch + 1) & 1][0]);

    // B operand (weights) from LDS: lane holds N=laneM(+16), its two K values.
    const float*    bp0 = &bsh[ch & 1][laneM * KC + 2 * laneH];
    const float*    bp1 = bp0 + 16 * KC;
    const unsigned* op  = &offs[ch * KC + 2 * laneH];

#pragma unroll
    for (int s = 0; s < STEPS; ++s) {
      const uint2 o2 = *(const uint2*)(op + 4 * s);        // ds_load_b64
      v2f a;                                               // saddr + u32 voffset:
      a.x = *(const float*)(xb + (abyte + o2.x));          //   v_add_nc_u32 +
      a.y = *(const float*)(xb + (abyte + o2.y));          //   global_load_b32
      const v2f b0 = *(const v2f*)(bp0 + 4 * s);           // ds_load_b64
      const v2f b1 = *(const v2f*)(bp1 + 4 * s);           // ds_load_b64
      acc0 = __builtin_amdgcn_wmma_f32_16x16x4_f32(
                 false, a, false, b0, (short)0, acc0, false, false);
      acc1 = __builtin_amdgcn_wmma_f32_16x16x4_f32(
                 false, a, false, b1, (short)0, acc1, false, false);
    }
    __syncthreads();   // everyone done with bsh[ch&1] before TDM reuses it
  }

  // Store + ReLU. D layout: VGPR v -> M = v + 8*laneH (consecutive d4), N = laneM.
  const long obase = (long)b * OSB + d1 * OS1 + d2 * OS2 + d3 * OS3;
  float* o0 = out + obase + (long)laneM * OSO + d4base + 8 * laneH;
  float* o1 = o0 + 16L * OSO;
#pragma unroll
  for (int v = 0; v < 8; ++v) {
    const int m = v + 8 * laneH;
    if (d4base + m < D4o) {        // divergence only after all WMMAs are done
      float r0 = acc0[v]; o0[v] = r0 > 0.f ? r0 : 0.f;
      float r1 = acc1[v]; o1[v] = r1 > 0.f ? r1 : 0.f;
    }
  }
}

extern "C" void kernel_launch(void* const* d_in, const int* in_sizes, int n_in,
                              void* d_out, int out_size, void* d_ws, size_t ws_size,
                              hipStream_t stream) {
  (void)in_sizes; (void)n_in; (void)d_ws; (void)ws_size; (void)out_size;
  const float* x = (const float*)d_in[0];   // pic_in  [2,16,8,8,64,64] fp32
  const float* w = (const float*)d_in[1];   // weight  [32,16,3,3,3,3]  fp32
  float* out = (float*)d_out;               // [2,32,6,6,62,62] fp32
  dim3 grid(ROWS / 2), block(256);          // 2232 workgroups x 8 waves
  conv4d_relu_wmma<<<grid, block, 0, stream>>>(x, w, out);
}